// AssociativeMemoryModule_43963285242298
// MI455X (gfx1250) — compile-verified
//
#include <hip/hip_runtime.h>
#include <math.h>

// ---------------------------------------------------------------------------
// Causal linear attention (elu+1 feature map) for MI455X / gfx1250.
// All matmul-shaped work goes through V_WMMA_F32_16X16X4_F32 (fp32 WMMA,
// matches fp32 reference precision).
//  - GEMMs: one wave per 16x64 output tile (4 accumulators) so each A
//    fragment feeds 4 WMMAs per K-step (load-issue bound otherwise).
//  - Attention: chunked (C=16) scan; per-chunk 16x16 WMMA tiles with the
//    running state S (64x64 per head) held in LDS.
// ---------------------------------------------------------------------------

typedef float v2f __attribute__((ext_vector_type(2)));
typedef float v8f __attribute__((ext_vector_type(8)));

#define HEADS    8
#define HEAD_DIM 64
#define DIM      512
#define SEQ      1024
#define BATCH    2
#define CH       16   // attention chunk = one WMMA tile in M
#define LSTR     65   // padded LDS row stride (bank-conflict free)

static __device__ __forceinline__ v8f wmma_f32(v2f a, v2f b, v8f c) {
    return __builtin_amdgcn_wmma_f32_16x16x4_f32(false, a, false, b,
                                                 (short)0, c, false, false);
}

// ---------------------------------------------------------------------------
// Kernel 1: fused QKV projection.  y = x @ W^T + bias, elu+1 on q/k, written
// in (b, h, t, d) layout.  One wave computes a 16x64 output tile (exactly one
// head's columns -> contiguous store in d).
// grid = (2048/16, 512/64, 3), block = 32 (one wave32)
// ---------------------------------------------------------------------------
__global__ __launch_bounds__(32)
void qkv_proj_kernel(const float* __restrict__ x,
                     const float* __restrict__ Wq, const float* __restrict__ bq,
                     const float* __restrict__ Wk, const float* __restrict__ bk,
                     const float* __restrict__ Wv, const float* __restrict__ bv,
                     float* __restrict__ qbuf, float* __restrict__ kbuf,
                     float* __restrict__ vbuf)
{
    const int m0    = blockIdx.x * 16;          // token-row tile
    const int n0    = blockIdx.y * 64;          // head-column tile (one head)
    const int which = blockIdx.z;               // 0=q, 1=k, 2=v

    const float* W    = (which == 0) ? Wq : (which == 1) ? Wk : Wv;
    const float* bias = (which == 0) ? bq : (which == 1) ? bk : bv;
    float*       dst  = (which == 0) ? qbuf : (which == 1) ? kbuf : vbuf;

    const int lane = threadIdx.x;
    const int half = lane >> 4;                 // 0: K+{0,1}, 1: K+{2,3}
    const int l16  = lane & 15;

    const float* arow = x + (size_t)(m0 + l16) * DIM;   // A: rows of x
    const float* brow = W + (size_t)(n0 + l16) * DIM;   // B[k][n] = W[n][k]

    v8f acc0 = {0.f,0.f,0.f,0.f,0.f,0.f,0.f,0.f};
    v8f acc1 = acc0, acc2 = acc0, acc3 = acc0;

    for (int k = 0; k < DIM; k += 4) {
        const int kk = k + half * 2;
        v2f a  = *(const v2f*)(arow + kk);                       // shared A frag
        v2f b0 = *(const v2f*)(brow + (size_t) 0 * DIM + kk);
        v2f b1 = *(const v2f*)(brow + (size_t)16 * DIM + kk);
        v2f b2 = *(const v2f*)(brow + (size_t)32 * DIM + kk);
        v2f b3 = *(const v2f*)(brow + (size_t)48 * DIM + kk);
        acc0 = wmma_f32(a, b0, acc0);
        acc1 = wmma_f32(a, b1, acc1);
        acc2 = wmma_f32(a, b2, acc2);
        acc3 = wmma_f32(a, b3, acc3);
    }

    const int h = n0 >> 6;                      // whole tile is one head
    v8f accs[4] = {acc0, acc1, acc2, acc3};
    for (int nt = 0; nt < 4; ++nt) {
        for (int g = 0; g < 8; ++g) {
            const int m = m0 + g + half * 8;    // C/D: VGPR g -> M=g+8*half
            const int n = n0 + nt * 16 + l16;
            float v = accs[nt][g] + bias[n];
            if (which < 2) v = (v > 0.f) ? (v + 1.f) : expf(v);   // elu(v)+1
            const int bb = m / SEQ, t = m % SEQ;
            const int d  = nt * 16 + l16;
            dst[(((size_t)(bb * HEADS + h) * SEQ) + t) * HEAD_DIM + d] = v;
        }
    }
}

// ---------------------------------------------------------------------------
// Kernel 2: chunked causal linear attention.  One workgroup (4 wave32) per
// (b, h) head; per chunk of 16 timesteps:
//   P = Q K^T (masked causal), out = Q*S_prev + tril(P)*V,
//   denom = Q*z_prev + rowsum(tril(P)), then S += K^T V, z += colsum(K).
// All matmuls are 16x16 fp32 WMMA tiles; S lives in LDS.
// grid = 16, block = 128
// ---------------------------------------------------------------------------
__global__ __launch_bounds__(128)
void linattn_kernel(const float* __restrict__ qbuf,
                    const float* __restrict__ kbuf,
                    const float* __restrict__ vbuf,
                    float* __restrict__ abuf)
{
    __shared__ float Ssm[HEAD_DIM * LSTR];   // running state S (64x64, padded)
    __shared__ float Qsm[CH * LSTR];
    __shared__ float Ksm[CH * LSTR];
    __shared__ float Vsm[CH * LSTR];
    __shared__ float Psm[CH * 17];           // masked Q K^T
    __shared__ float zsm[HEAD_DIM];          // running k-feature sum
    __shared__ float dsm[CH];                // per-row denominator

    const int bh   = blockIdx.x;
    const int tid  = threadIdx.x;
    const int wave = tid >> 5;
    const int lane = tid & 31;
    const int half = lane >> 4;
    const int l16  = lane & 15;
    const int nc0  = wave * 16;              // this wave's head-dim columns

    const float* Qb = qbuf + (size_t)bh * SEQ * HEAD_DIM;
    const float* Kb = kbuf + (size_t)bh * SEQ * HEAD_DIM;
    const float* Vb = vbuf + (size_t)bh * SEQ * HEAD_DIM;
    float*       Ob = abuf + (size_t)bh * SEQ * HEAD_DIM;

    for (int i = tid; i < HEAD_DIM * LSTR; i += 128) Ssm[i] = 0.f;
    for (int i = tid; i < HEAD_DIM; i += 128)        zsm[i] = 0.f;
    __syncthreads();

    for (int t0 = 0; t0 < SEQ; t0 += CH) {
        // ---- stage chunk tiles (coalesced) -------------------------------
        for (int i = tid; i < CH * HEAD_DIM; i += 128) {
            const int r = i >> 6, c = i & 63;
            Qsm[r * LSTR + c] = Qb[(size_t)(t0 + r) * HEAD_DIM + c];
            Ksm[r * LSTR + c] = Kb[(size_t)(t0 + r) * HEAD_DIM + c];
            Vsm[r * LSTR + c] = Vb[(size_t)(t0 + r) * HEAD_DIM + c];
        }
        __syncthreads();

        // ---- wave 0: P = Q K^T, causal-masked into LDS -------------------
        if (wave == 0) {
            v8f p = {0.f,0.f,0.f,0.f,0.f,0.f,0.f,0.f};
            for (int d = 0; d < HEAD_DIM; d += 4) {
                const int dd = d + half * 2;
                v2f a = { Qsm[l16 * LSTR + dd], Qsm[l16 * LSTR + dd + 1] };
                v2f b = { Ksm[l16 * LSTR + dd], Ksm[l16 * LSTR + dd + 1] };
                p = wmma_f32(a, b, p);
            }
            for (int g = 0; g < 8; ++g) {
                const int i = g + half * 8;      // row
                Psm[i * 17 + l16] = (l16 <= i) ? p[g] : 0.f;
            }
        }
        __syncthreads();

        // ---- denominators: q . z_prev + rowsum(tril(P)) ------------------
        if (tid < CH) {
            float dsum = 0.f;
            for (int d = 0; d < HEAD_DIM; ++d) dsum += Qsm[tid * LSTR + d] * zsm[d];
            for (int j = 0; j <= tid; ++j)     dsum += Psm[tid * 17 + j];
            dsm[tid] = fmaxf(dsum, 1e-6f);
        }

        // ---- out = Q @ S_prev  +  tril(P) @ V  (this wave's 16 cols) -----
        v8f o = {0.f,0.f,0.f,0.f,0.f,0.f,0.f,0.f};
        for (int d = 0; d < HEAD_DIM; d += 4) {
            const int dd = d + half * 2;
            v2f a = { Qsm[l16 * LSTR + dd], Qsm[l16 * LSTR + dd + 1] };
            v2f b = { Ssm[dd * LSTR + nc0 + l16], Ssm[(dd + 1) * LSTR + nc0 + l16] };
            o = wmma_f32(a, b, o);
        }
        for (int j = 0; j < CH; j += 4) {
            const int jj = j + half * 2;
            v2f a = { Psm[l16 * 17 + jj], Psm[l16 * 17 + jj + 1] };
            v2f b = { Vsm[jj * LSTR + nc0 + l16], Vsm[(jj + 1) * LSTR + nc0 + l16] };
            o = wmma_f32(a, b, o);
        }
        __syncthreads();   // S/z/P/denom reads done before state update

        // ---- state update: S[:, nc0:nc0+16] += K^T @ V -------------------
        for (int mt = 0; mt < 4; ++mt) {
            v8f sacc = {0.f,0.f,0.f,0.f,0.f,0.f,0.f,0.f};
            for (int j = 0; j < CH; j += 4) {
                const int jj = j + half * 2;
                // A[i][k] = (K^T)[16*mt+i][k] = K[k][16*mt+i]
                v2f a = { Ksm[jj * LSTR + mt * 16 + l16],
                          Ksm[(jj + 1) * LSTR + mt * 16 + l16] };
                v2f b = { Vsm[jj * LSTR + nc0 + l16],
                          Vsm[(jj + 1) * LSTR + nc0 + l16] };
                sacc = wmma_f32(a, b, sacc);
            }
            for (int g = 0; g < 8; ++g) {
                const int m = mt * 16 + g + half * 8;
                Ssm[m * LSTR + nc0 + l16] += sacc[g];
            }
        }
        // z += colsum(K)  (after denominators used z_prev)
        if (tid < HEAD_DIM) {
            float s = 0.f;
            for (int j = 0; j < CH; ++j) s += Ksm[j * LSTR + tid];
            zsm[tid] += s;
        }

        // ---- write normalized output ------------------------------------
        for (int g = 0; g < 8; ++g) {
            const int i = g + half * 8;
            Ob[(size_t)(t0 + i) * HEAD_DIM + nc0 + l16] = o[g] / dsm[i];
        }
        __syncthreads();   // protect K/V/S/z vs next chunk's staging
    }
}

// ---------------------------------------------------------------------------
// Kernel 3: output projection.  out = attn @ Wo^T + bo (gating weight = 1.0).
// One wave per 16x64 output tile; A-fragments gather from (b,h,t,d) layout.
// grid = (2048/16, 512/64), block = 32
// ---------------------------------------------------------------------------
__global__ __launch_bounds__(32)
void out_proj_kernel(const float* __restrict__ abuf,
                     const float* __restrict__ Wo,
                     const float* __restrict__ bo,
                     float* __restrict__ out)
{
    const int m0   = blockIdx.x * 16;
    const int n0   = blockIdx.y * 64;
    const int lane = threadIdx.x;
    const int half = lane >> 4;
    const int l16  = lane & 15;

    const int r  = m0 + l16;                 // token row for A fragment
    const int bb = r / SEQ, t = r % SEQ;
    const float* brow = Wo + (size_t)(n0 + l16) * DIM;

    v8f acc0 = {0.f,0.f,0.f,0.f,0.f,0.f,0.f,0.f};
    v8f acc1 = acc0, acc2 = acc0, acc3 = acc0;

    for (int k = 0; k < DIM; k += 4) {
        const int kk = k + half * 2;
        const int h  = kk >> 6, d = kk & 63;     // kk even -> d, d+1 same head
        v2f a  = *(const v2f*)(abuf +
                  ((size_t)(bb * HEADS + h) * SEQ + t) * HEAD_DIM + d);
        v2f b0 = *(const v2f*)(brow + (size_t) 0 * DIM + kk);
        v2f b1 = *(const v2f*)(brow + (size_t)16 * DIM + kk);
        v2f b2 = *(const v2f*)(brow + (size_t)32 * DIM + kk);
        v2f b3 = *(const v2f*)(brow + (size_t)48 * DIM + kk);
        acc0 = wmma_f32(a, b0, acc0);
        acc1 = wmma_f32(a, b1, acc1);
        acc2 = wmma_f32(a, b2, acc2);
        acc3 = wmma_f32(a, b3, acc3);
    }

    v8f accs[4] = {acc0, acc1, acc2, acc3};
    for (int nt = 0; nt < 4; ++nt) {
        for (int g = 0; g < 8; ++g) {
            const int m = m0 + g + half * 8;
            const int n = n0 + nt * 16 + l16;
            out[(size_t)m * DIM + n] = (accs[nt][g] + bo[n]) * 1.0f;  // gating
        }
    }
}

// ---------------------------------------------------------------------------
extern "C" void kernel_launch(void* const* d_in, const int* in_sizes, int n_in,
                              void* d_out, int out_size, void* d_ws, size_t ws_size,
                              hipStream_t stream)
{
    (void)in_sizes; (void)n_in; (void)out_size; (void)ws_size;

    const float* x  = (const float*)d_in[0];
    const float* Wq = (const float*)d_in[1];
    const float* bq = (const float*)d_in[2];
    const float* Wk = (const float*)d_in[3];
    const float* bk = (const float*)d_in[4];
    const float* Wv = (const float*)d_in[5];
    const float* bv = (const float*)d_in[6];
    const float* Wo = (const float*)d_in[7];
    const float* bo = (const float*)d_in[8];
    float* out = (float*)d_out;

    const size_t NTOK = (size_t)BATCH * SEQ;          // 2048
    float* qbuf = (float*)d_ws;                       // (b,h,t,d) 4 MB
    float* kbuf = qbuf + NTOK * DIM;
    float* vbuf = kbuf + NTOK * DIM;
    float* abuf = vbuf + NTOK * DIM;                  // attention output

    dim3 gproj(NTOK / 16, DIM / 64, 3);
    qkv_proj_kernel<<<gproj, 32, 0, stream>>>(x, Wq, bq, Wk, bk, Wv, bv,
                                              qbuf, kbuf, vbuf);

    linattn_kernel<<<dim3(BATCH * HEADS), 128, 0, stream>>>(qbuf, kbuf, vbuf, abuf);

    out_proj_kernel<<<dim3(NTOK / 16, DIM / 64), 32, 0, stream>>>(abuf, Wo, bo, out);
}